// Model_18124761989625
// MI455X (gfx1250) — compile-verified
//
#include <hip/hip_runtime.h>
#include <hip/hip_bf16.h>

// ---------------- problem constants (from reference) ----------------
#define N_NODES 20000
#define N_EDGES 240000
#define IN_DIM  20
#define SH_DIM  16
#define OUT_DIM 360          // 20 (mlp out) + 340 (aggr)
#define N_TILES (N_NODES / 16)   // 1250, exact

typedef __attribute__((ext_vector_type(2))) float v2f;
typedef __attribute__((ext_vector_type(8))) float v8f;

// ---------------- kernel 0: zero the output accumulator ----------------
__global__ void zero_out_kernel(float* __restrict__ out, int n) {
    int idx = blockIdx.x * blockDim.x + threadIdx.x;
    if (idx < n) out[idx] = 0.0f;
}

// ---------------- kernel 1: node MLP via fp32 WMMA ----------------
// D = A(16xK) x B(Kx16): B[k][n] = W[n][k]  (computes x @ W^T)
// Branch-free padding: unconditional load from a clamped in-bounds address,
// then select-to-zero (v_cndmask) for the pad columns — no EXEC save/restore,
// so B loads can be clustered ahead of the WMMA chain.
__device__ __forceinline__ v2f load_bfrag(const float* __restrict__ W,
                                          int ntile, int kt, int lane) {
    int n = ntile * 16 + (lane & 15);
    int k = kt * 4 + (lane >> 4) * 2;
    bool valid = (n < IN_DIM);
    int nn = valid ? n : 0;                  // clamped, always in-bounds
    float bx = W[nn * IN_DIM + k];
    float by = W[nn * IN_DIM + k + 1];
    v2f b;
    b.x = valid ? bx : 0.0f;
    b.y = valid ? by : 0.0f;
    return b;
}

__global__ __launch_bounds__(128) void node_mlp_kernel(
    const float* __restrict__ x,
    const float* __restrict__ W_pre,  const float* __restrict__ b_pre,
    const float* __restrict__ W_post, const float* __restrict__ b_post,
    const float* __restrict__ W_sc,   const float* __restrict__ b_sc,
    float* __restrict__ out)
{
    __shared__ float lds_h[4][16][36];   // per-wave 16x20 relu(h1) tile (padded)

    const int lane  = threadIdx.x & 31;
    const int wave  = threadIdx.x >> 5;
    const int tile  = blockIdx.x * 4 + wave;
    if (tile >= N_TILES) return;         // wave-uniform guard: EXEC stays all-1s

    const int row0  = tile * 16;
    const int m     = lane & 15;         // A: row within tile; D: column
    const int khalf = lane >> 4;         // selects K pair (0-1 vs 2-3)

    // ---- A fragments of the x tile (reused for W_pre and W_sc GEMMs) ----
    v2f ax[5];
#pragma unroll
    for (int kt = 0; kt < 5; ++kt) {
        const float* p = x + (size_t)(row0 + m) * IN_DIM + kt * 4 + khalf * 2;
        ax[kt].x = p[0];
        ax[kt].y = p[1];
    }

    // ---- GEMM1: h1 = relu(x @ W_pre^T + b_pre), stage into LDS ----
#pragma unroll
    for (int ntile = 0; ntile < 2; ++ntile) {
        v2f bf[5];
#pragma unroll
        for (int kt = 0; kt < 5; ++kt) bf[kt] = load_bfrag(W_pre, ntile, kt, lane);
        v8f acc = {};
#pragma unroll
        for (int kt = 0; kt < 5; ++kt) {
            acc = __builtin_amdgcn_wmma_f32_16x16x4_f32(
                false, ax[kt], false, bf[kt], (short)0, acc, false, false);
        }
        int nc = ntile * 16 + m;
        if (nc < IN_DIM) {
            float bias = b_pre[nc];
#pragma unroll
            for (int v = 0; v < 8; ++v) {
                int mr = v + khalf * 8;
                lds_h[wave][mr][nc] = fmaxf(acc[v] + bias, 0.0f);
            }
        }
    }

    // Same-wave DS ops are in-order; drain the DS counter before consuming.
    asm volatile("s_wait_dscnt 0" ::: "memory");

    // ---- reload h1 in A-matrix layout ----
    v2f ah[5];
#pragma unroll
    for (int kt = 0; kt < 5; ++kt) {
        int k = kt * 4 + khalf * 2;
        ah[kt].x = lds_h[wave][m][k];
        ah[kt].y = lds_h[wave][m][k + 1];
    }
    asm volatile("s_wait_dscnt 0" ::: "memory");

    // ---- GEMM2+3 chained through C:  out = x@W_sc^T + h1@W_post^T (+ biases) ----
#pragma unroll
    for (int ntile = 0; ntile < 2; ++ntile) {
        v2f bsc[5], bpo[5];
#pragma unroll
        for (int kt = 0; kt < 5; ++kt) bsc[kt] = load_bfrag(W_sc, ntile, kt, lane);
#pragma unroll
        for (int kt = 0; kt < 5; ++kt) bpo[kt] = load_bfrag(W_post, ntile, kt, lane);
        v8f acc = {};
#pragma unroll
        for (int kt = 0; kt < 5; ++kt) {
            acc = __builtin_amdgcn_wmma_f32_16x16x4_f32(
                false, ax[kt], false, bsc[kt], (short)0, acc, false, false);
        }
#pragma unroll
        for (int kt = 0; kt < 5; ++kt) {
            acc = __builtin_amdgcn_wmma_f32_16x16x4_f32(
                false, ah[kt], false, bpo[kt], (short)0, acc, false, false);
        }
        int nc = ntile * 16 + m;
        if (nc < IN_DIM) {
            float bias = b_sc[nc] + b_post[nc];
#pragma unroll
            for (int v = 0; v < 8; ++v) {
                int mr = v + khalf * 8;
                out[(size_t)(row0 + mr) * OUT_DIM + nc] = acc[v] + bias;
            }
        }
    }
}

// ---------------- kernel 2: edge SH outer-product scatter ----------------
// One thread per (edge, i) pair, i in [0,20): 1 sender atomic + 16 tp atomics.
__global__ __launch_bounds__(256) void edge_scatter_kernel(
    const float* __restrict__ x,
    const float* __restrict__ pos,
    const int*   __restrict__ edge_index,
    float* __restrict__ out)
{
    long long tid = (long long)blockIdx.x * blockDim.x + threadIdx.x;
    const long long total = (long long)N_EDGES * IN_DIM;
    if (tid >= total) return;
    int e = (int)(tid / IN_DIM);
    int i = (int)(tid % IN_DIM);

    int r = edge_index[e];
    int c = edge_index[N_EDGES + e];

    float px = pos[c * 3 + 0] - pos[r * 3 + 0];
    float py = pos[c * 3 + 1] - pos[r * 3 + 1];
    float pz = pos[c * 3 + 2] - pos[r * 3 + 2];
    float inv = 1.0f / sqrtf(px * px + py * py + pz * pz + 1e-12f);
    float X = px * inv, Y = py * inv, Z = pz * inv;
    float X2 = X * X, Y2 = Y * Y, Z2 = Z * Z;

    float sh[SH_DIM];
    sh[0]  = 0.28209479177387814f;
    sh[1]  = 0.4886025119029199f * Y;
    sh[2]  = 0.4886025119029199f * Z;
    sh[3]  = 0.4886025119029199f * X;
    sh[4]  = 1.0925484305920792f * X * Y;
    sh[5]  = 1.0925484305920792f * Y * Z;
    sh[6]  = 0.31539156525252005f * (3.0f * Z2 - 1.0f);
    sh[7]  = 1.0925484305920792f * X * Z;
    sh[8]  = 0.5462742152960396f * (X2 - Y2);
    sh[9]  = 0.5900435899266435f * Y * (3.0f * X2 - Y2);
    sh[10] = 2.890611442640554f * X * Y * Z;
    sh[11] = 0.4570457994644658f * Y * (5.0f * Z2 - 1.0f);
    sh[12] = 0.3731763325901154f * Z * (5.0f * Z2 - 3.0f);
    sh[13] = 0.4570457994644658f * X * (5.0f * Z2 - 1.0f);
    sh[14] = 1.445305721320277f * Z * (X2 - Y2);
    sh[15] = 0.5900435899266435f * X * (X2 - 3.0f * Y2);

    float s = x[(size_t)r * IN_DIM + i];
    float* dst = out + (size_t)c * OUT_DIM;

    atomicAdd(dst + IN_DIM + i, s);                 // sender part (cols 20..39)
#pragma unroll
    for (int j = 0; j < SH_DIM; ++j)                // tp part (cols 40..359)
        atomicAdd(dst + 2 * IN_DIM + i * SH_DIM + j, s * sh[j]);
}

// ---------------- launcher ----------------
extern "C" void kernel_launch(void* const* d_in, const int* in_sizes, int n_in,
                              void* d_out, int out_size, void* d_ws, size_t ws_size,
                              hipStream_t stream) {
    const float* x          = (const float*)d_in[0];
    const float* positions  = (const float*)d_in[1];
    const int*   edge_index = (const int*)  d_in[2];
    const float* W_pre      = (const float*)d_in[3];
    const float* b_pre      = (const float*)d_in[4];
    const float* W_post     = (const float*)d_in[5];
    const float* b_post     = (const float*)d_in[6];
    const float* W_sc       = (const float*)d_in[7];
    const float* b_sc       = (const float*)d_in[8];
    float* out = (float*)d_out;

    // 1) zero output (edge kernel accumulates with atomics; d_out not re-zeroed by harness)
    zero_out_kernel<<<(out_size + 255) / 256, 256, 0, stream>>>(out, out_size);

    // 2) node MLP, 4 waves/block, 1 wave per 16-row tile
    node_mlp_kernel<<<(N_TILES + 3) / 4, 128, 0, stream>>>(
        x, W_pre, b_pre, W_post, b_post, W_sc, b_sc, out);

    // 3) edge scatter: 240000 edges x 20 feature rows
    long long total = (long long)N_EDGES * IN_DIM;
    edge_scatter_kernel<<<(int)((total + 255) / 256), 256, 0, stream>>>(
        x, positions, edge_index, out);
}